// Gate_20976620274020
// MI455X (gfx1250) — compile-verified
//
#include <hip/hip_runtime.h>
#include <hip/hip_bf16.h>

typedef __attribute__((ext_vector_type(2))) float v2f;
typedef __attribute__((ext_vector_type(8))) float v8f;

namespace {
constexpr int kD             = 2048;          // hidden dim
constexpr int kE             = 64;            // experts
constexpr int kTokens        = 4 * 4096;      // B*S
constexpr int kTokensPerWave = 16;            // one WMMA M-tile
constexpr int kWavesPerBlock = 8;             // 256 threads
constexpr int kLdsStride     = 65;            // 64 + 1 pad: conflict-free row reads
}

__global__ __launch_bounds__(kWavesPerBlock * 32)
void moe_gate_topk_kernel(const float* __restrict__ x,
                          const float* __restrict__ W,
                          const float* __restrict__ bias,
                          float* __restrict__ gate_out,
                          int* __restrict__ idx_out)
{
  __shared__ float lds[kWavesPerBlock][16 * kLdsStride];

  const int lane = threadIdx.x & 31;
  const int wave = threadIdx.x >> 5;
  const int tile = blockIdx.x * kWavesPerBlock + wave;   // 16 tokens per tile
  const int r16  = lane & 15;
  const int hi   = lane >> 4;                            // 0: K=0,1  1: K=2,3

  // A: x rows (tokens); B: W rows (experts). Both feed WMMA as float2/lane.
  const float* aptr = x + (size_t)tile * kTokensPerWave * kD + (size_t)r16 * kD + 2 * hi;
  const float* bptr = W + (size_t)r16 * kD + 2 * hi;

  v8f acc0 = {}; v8f acc1 = {}; v8f acc2 = {}; v8f acc3 = {};

  for (int kk = 0; kk < kD; kk += 64) {
    __builtin_prefetch(aptr + kk + 512, 0, 0);   // stream x ahead (global_prefetch_b8)
    #pragma unroll
    for (int k4 = 0; k4 < 64; k4 += 4) {
      const int k = kk + k4;
      v2f a  = *(const v2f*)(aptr + k);
      v2f b0 = *(const v2f*)(bptr + 0 * 16 * kD + k);
      v2f b1 = *(const v2f*)(bptr + 1 * 16 * kD + k);
      v2f b2 = *(const v2f*)(bptr + 2 * 16 * kD + k);
      v2f b3 = *(const v2f*)(bptr + 3 * 16 * kD + k);
      acc0 = __builtin_amdgcn_wmma_f32_16x16x4_f32(false, a, false, b0, (short)0, acc0, false, false);
      acc1 = __builtin_amdgcn_wmma_f32_16x16x4_f32(false, a, false, b1, (short)0, acc1, false, false);
      acc2 = __builtin_amdgcn_wmma_f32_16x16x4_f32(false, a, false, b2, (short)0, acc2, false, false);
      acc3 = __builtin_amdgcn_wmma_f32_16x16x4_f32(false, a, false, b3, (short)0, acc3, false, false);
    }
  }

  // Bias: lane's column is expert e = 16*t + r16 for every accumulator row.
  const float bv0 = bias[ 0 + r16];
  const float bv1 = bias[16 + r16];
  const float bv2 = bias[32 + r16];
  const float bv3 = bias[48 + r16];
  #pragma unroll
  for (int r = 0; r < 8; ++r) {
    acc0[r] += bv0; acc1[r] += bv1; acc2[r] += bv2; acc3[r] += bv3;
  }

  // Scatter 16x64 logit tile into padded LDS. C/D layout: VGPR r holds
  // M=r (lanes 0-15) and M=r+8 (lanes 16-31), N = lane%16 within each N-tile.
  float* my = lds[wave];
  #pragma unroll
  for (int r = 0; r < 8; ++r) {
    const int row = r + 8 * hi;
    my[row * kLdsStride +  0 + r16] = acc0[r];
    my[row * kLdsStride + 16 + r16] = acc1[r];
    my[row * kLdsStride + 32 + r16] = acc2[r];
    my[row * kLdsStride + 48 + r16] = acc3[r];
  }
  __syncthreads();

  // Top-2 scan per token: lanes 0..15, one token each. stride-65 => bank-free.
  float v0 = -3.402823466e+38f, v1 = -3.402823466e+38f;
  int i0 = 0, i1 = 0;
  if (lane < 16) {
    const float* row = my + lane * kLdsStride;
    #pragma unroll 8
    for (int j = 0; j < kE; ++j) {
      float v = row[j];
      if (v > v0)      { v1 = v0; i1 = i0; v0 = v; i0 = j; }
      else if (v > v1) { v1 = v;  i1 = j; }
    }
  }
  __syncthreads();

  // Zero the tile cooperatively.
  #pragma unroll
  for (int i = 0; i < 32; ++i) {
    const int idx = i * 32 + lane;
    my[(idx >> 6) * kLdsStride + (idx & 63)] = 0.0f;
  }
  __syncthreads();

  // Sparse softmax over the two surviving logits (others are exp(-inf)=0).
  if (lane < 16) {
    const float e1    = __expf(v1 - v0);
    const float denom = 1.0f + e1;
    float* row = my + lane * kLdsStride;
    row[i0] = 1.0f / denom;
    row[i1] = e1 / denom;
    const int token = tile * kTokensPerWave + lane;
    idx_out[2 * token + 0] = i0;
    idx_out[2 * token + 1] = i1;
  }
  __syncthreads();

  // Coalesced 16x64 writeback.
  float* outBase = gate_out + (size_t)tile * kTokensPerWave * kE;
  #pragma unroll
  for (int i = 0; i < 32; ++i) {
    const int idx = i * 32 + lane;
    outBase[idx] = my[(idx >> 6) * kLdsStride + (idx & 63)];
  }
}

extern "C" void kernel_launch(void* const* d_in, const int* in_sizes, int n_in,
                              void* d_out, int out_size, void* d_ws, size_t ws_size,
                              hipStream_t stream) {
  (void)in_sizes; (void)n_in; (void)out_size; (void)d_ws; (void)ws_size;
  const float* x = (const float*)d_in[0];
  const float* W = (const float*)d_in[1];
  const float* b = (const float*)d_in[2];
  float* gate = (float*)d_out;                                    // 16384*64 f32
  int*   idx  = (int*)((float*)d_out + (size_t)kTokens * kE);     // then 16384*2 i32

  dim3 grid(kTokens / (kTokensPerWave * kWavesPerBlock));         // 128
  dim3 block(kWavesPerBlock * 32);                                // 256
  hipLaunchKernelGGL(moe_gate_topk_kernel, grid, block, 0, stream, x, W, b, gate, idx);
}